// PermutationLayer_39290360824036
// MI455X (gfx1250) — compile-verified
//
#include <hip/hip_runtime.h>

typedef __attribute__((ext_vector_type(2))) float v2f;
typedef __attribute__((ext_vector_type(4))) float v4f;
typedef __attribute__((ext_vector_type(8))) float v8f;

#define NF 256      // permutation size f (matrix is NF x NF)
#define NR 256      // repetitions r
#define NN 256      // N = B*L*C = 16*4*4
#define PITCH 257   // LDS row pitch (bank-conflict-free: gcd(257,64)=1)
#define SINKHORN_ITERS 20

// ---------------------------------------------------------------------------
// Kernel A: X (B,F,L,C,R) -> WMMA-ready packed Xp.
//   X linear index  = (n>>4)<<20 | f<<12 | (n&15)<<8 | r   (contiguous in r)
//   Xp linear index = ((((r*64 + q)*2 + h)*256 + n)*2 + j), k = 4q + 2h + j
// so a WMMA lane's two K-values are adjacent -> one global_load_b64, and the
// 16 lanes of each half read a contiguous 128B segment.
// Each block handles an f-pair so packed stores are contiguous float2s.
// ---------------------------------------------------------------------------
__global__ __launch_bounds__(256) void xpose_in(const float* __restrict__ X,
                                                float* __restrict__ Xp) {
  __shared__ float tile[2][32][33];
  const int f0 = blockIdx.x * 2;  // even f; pair (f0, f0+1)
  const int nb = blockIdx.y * 32;
  const int rb = blockIdx.z * 32;
  const int x  = threadIdx.x;
  for (int yy = threadIdx.y; yy < 32; yy += 8) {
    const int n = nb + yy;
    const int r = rb + x;
    const size_t base =
        ((size_t)(n >> 4) << 20) + ((size_t)(n & 15) << 8) + r;
    tile[0][yy][x] = X[base + ((size_t)f0 << 12)];        // coalesced in r
    tile[1][yy][x] = X[base + ((size_t)(f0 + 1) << 12)];
  }
  __syncthreads();
  const int q = f0 >> 2;
  const int h = (f0 >> 1) & 1;
  for (int yy = threadIdx.y; yy < 32; yy += 8) {
    const int r = rb + yy;
    const int n = nb + x;
    v2f val;
    val.x = tile[0][x][yy];
    val.y = tile[1][x][yy];
    *(v2f*)(Xp + ((((size_t)r * 64 + q) * 2 + h) * 256 + n) * 2) = val;
  }
}

// ---------------------------------------------------------------------------
// Kernel B: per-r Sinkhorn (20 iters, log-space, LDS-resident 256x256) then
// D[m,n] = sum_f p[f,m] * X_r[f,n] via V_WMMA_F32_16X16X4_F32.
//   USE_XP : B-fragments via packed Xp (one b64 per lane) vs raw X (scalar).
//   USE_WS2: coalesced store of [r][m][n] to workspace (transposed later)
//            vs scattered direct store into the (b,m,l,c,r) output.
// One workgroup (256 thr = 8 waves) per r.
// Dynamic LDS: 256*257 matrix + 256 row-lse floats = 264192 B (< 320KB WGP).
// ---------------------------------------------------------------------------
template <bool USE_XP, bool USE_WS2>
__global__ __launch_bounds__(256) void sinkhorn_wmma(
    const float* __restrict__ seed,   // (F,F,R)
    const float* __restrict__ gum,    // (R,F,F)
    const float* __restrict__ Xsrc,   // packed Xp or raw X
    float* __restrict__ dst) {        // ws2[r][m][n] or final output
  extern __shared__ float sLA[];      // NF*PITCH matrix, then NF row-lse
  float* lseRow = sLA + NF * PITCH;
  const int r   = blockIdx.x;
  const int tid = threadIdx.x;

  // ---- Phase 1: log_alpha[f][m] = seed[f,m,r] + gumbel[r,f,m] ----
  for (int i = tid * 4; i < NF * NF; i += 256 * 4) {
    const int f = i >> 8, m = i & 255;
    const v4f g = *(const v4f*)(gum + ((size_t)r << 16) + i);  // 16B coalesced
    float* dstRow = sLA + f * PITCH + m;
#pragma unroll
    for (int j = 0; j < 4; ++j)
      dstRow[j] = seed[((size_t)(i + j) << 8) + r] + g[j];
  }
  __syncthreads();

  // ---- Phase 2: 20 Sinkhorn iterations ----
  // Row pass only computes lseRow (no write-back); column pass applies both
  // the row and column normalization in a single write (5R+1W per element
  // per iteration instead of 6R+2W).
  for (int it = 0; it < SINKHORN_ITERS; ++it) {
    {  // row lse: thread tid owns row f = tid
      const float* row = sLA + tid * PITCH;
      float mx = -__builtin_inff();
      for (int m = 0; m < NF; ++m) mx = fmaxf(mx, row[m]);
      float s = 0.0f;
      for (int m = 0; m < NF; ++m) s += __expf(row[m] - mx);
      lseRow[tid] = mx + __logf(s);
    }
    __syncthreads();
    {  // column: thread tid owns column m = tid; lseRow reads broadcast
      float* col = sLA + tid;
      float mx = -__builtin_inff();
      for (int f = 0; f < NF; ++f)
        mx = fmaxf(mx, col[f * PITCH] - lseRow[f]);
      float s = 0.0f;
      for (int f = 0; f < NF; ++f)
        s += __expf(col[f * PITCH] - lseRow[f] - mx);
      const float lse = mx + __logf(s);
      for (int f = 0; f < NF; ++f)
        col[f * PITCH] = col[f * PITCH] - lseRow[f] - lse;
    }
    __syncthreads();
  }

  // ---- Phase 3: p = exp(log_alpha), in place ----
  for (int i = tid; i < NF * NF; i += 256) {
    const int f = i >> 8, m = i & 255;
    sLA[f * PITCH + m] = __expf(sLA[f * PITCH + m]);
  }
  __syncthreads();

  // ---- Phase 4: D = p^T * X_r with 16x16x4 f32 WMMA ----
  // A[m,k] = p[k][m] from LDS (ds_load_2addr pairs); B[k,n] from global.
  const int lane   = tid & 31;
  const int wave   = tid >> 5;
  const int hiHalf = lane >> 4;   // 0: K pair {k0,k0+1}; 1: {k0+2,k0+3}
  const int lane16 = lane & 15;

  for (int tn = wave; tn < 16; tn += 8) {     // wave-uniform: EXEC all-ones
    const int n = tn * 16 + lane16;
    for (int tmq = 0; tmq < 4; ++tmq) {
      const int tm0 = tmq * 4;
      v8f acc[4];
#pragma unroll
      for (int j = 0; j < 4; ++j)
#pragma unroll
        for (int v = 0; v < 8; ++v) acc[j][v] = 0.0f;

      for (int k0 = 0; k0 < NF; k0 += 4) {
        const int kb = k0 + hiHalf * 2;
        v2f bfrag;
        if (USE_XP) {
          // one 8-byte load; contiguous 128B per 16-lane half
          bfrag = *(const v2f*)(Xsrc +
              ((((size_t)r * 64 + (k0 >> 2)) * 2 + hiHalf) * 256 + n) * 2);
        } else {  // raw X: (n>>4)<<20 | f<<12 | (n&15)<<8 | r
          const size_t base =
              ((size_t)(n >> 4) << 20) + ((size_t)(n & 15) << 8) + r;
          bfrag.x = Xsrc[base + ((size_t)kb << 12)];
          bfrag.y = Xsrc[base + ((size_t)(kb + 1) << 12)];
        }
#pragma unroll
        for (int j = 0; j < 4; ++j) {
          const int m = (tm0 + j) * 16 + lane16;
          v2f afrag;
          const float* ap = sLA + kb * PITCH + m;
          afrag.x = ap[0];
          afrag.y = ap[PITCH];
          acc[j] = __builtin_amdgcn_wmma_f32_16x16x4_f32(
              false, afrag, false, bfrag, (short)0, acc[j], false, false);
        }
      }
      // store D tiles. C/D layout: lanes 0-15: M=v, N=lane; 16-31: M=v+8.
#pragma unroll
      for (int j = 0; j < 4; ++j) {
        const int mBase = (tm0 + j) * 16 + hiHalf * 8;
#pragma unroll
        for (int v = 0; v < 8; ++v) {
          const int M = mBase + v;
          const float val = acc[j][v];
          if (USE_WS2) {
            dst[((size_t)r << 16) + ((size_t)M << 8) + n] = val;
          } else {  // out[b,m,l,c,r] = (n>>4)<<20 | M<<12 | (n&15)<<8 | r
            dst[((size_t)(n >> 4) << 20) + ((size_t)M << 12) +
                ((size_t)(n & 15) << 8) + r] = val;
          }
        }
      }
    }
  }
}

// ---------------------------------------------------------------------------
// Kernel C: ws2[r][m][n] -> out (b,m,l,c,r). Same index algebra as X:
// out index = (n>>4)<<20 | m<<12 | (n&15)<<8 | r.
// Loads coalesced over n, stores coalesced over r.
// ---------------------------------------------------------------------------
__global__ __launch_bounds__(256) void xpose_out(const float* __restrict__ ws2,
                                                 float* __restrict__ out) {
  __shared__ float tile[32][33];
  const int m  = blockIdx.x;
  const int nb = blockIdx.y * 32;
  const int rb = blockIdx.z * 32;
  const int x  = threadIdx.x;
  for (int yy = threadIdx.y; yy < 32; yy += 8) {
    const int r = rb + yy;
    const int n = nb + x;
    tile[yy][x] = ws2[((size_t)r << 16) + ((size_t)m << 8) + n];
  }
  __syncthreads();
  for (int yy = threadIdx.y; yy < 32; yy += 8) {
    const int n = nb + yy;
    const int r = rb + x;
    out[((size_t)(n >> 4) << 20) + ((size_t)m << 12) +
        ((size_t)(n & 15) << 8) + r] = tile[x][yy];
  }
}

// ---------------------------------------------------------------------------
extern "C" void kernel_launch(void* const* d_in, const int* in_sizes, int n_in,
                              void* d_out, int out_size, void* d_ws,
                              size_t ws_size, hipStream_t stream) {
  (void)in_sizes; (void)n_in; (void)out_size;
  const float* X    = (const float*)d_in[0];  // (B,F,L,C,R)
  const float* seed = (const float*)d_in[1];  // (F,F,R)
  const float* gum  = (const float*)d_in[2];  // (R,F,F)
  float* out = (float*)d_out;

  const size_t bytesXp = (size_t)NR * NF * NN * sizeof(float);        // 64 MiB
  const size_t bytesW2 = bytesXp;                                     // 64 MiB
  const size_t smem = ((size_t)NF * PITCH + NF) * sizeof(float);      // 264192

  const bool haveXp = ws_size >= bytesXp;
  const bool haveW2 = ws_size >= bytesXp + bytesW2;
  float* Xp  = (float*)d_ws;
  float* ws2 = (float*)((char*)d_ws + bytesXp);

  // Allow >64KB dynamic LDS for the Sinkhorn kernel (CDNA5 WGP: 320KB).
  (void)hipFuncSetAttribute(
      reinterpret_cast<const void*>(&sinkhorn_wmma<true, true>),
      hipFuncAttributeMaxDynamicSharedMemorySize, (int)smem);
  (void)hipFuncSetAttribute(
      reinterpret_cast<const void*>(&sinkhorn_wmma<true, false>),
      hipFuncAttributeMaxDynamicSharedMemorySize, (int)smem);
  (void)hipFuncSetAttribute(
      reinterpret_cast<const void*>(&sinkhorn_wmma<false, false>),
      hipFuncAttributeMaxDynamicSharedMemorySize, (int)smem);

  const dim3 tblk(32, 8);

  if (haveXp) {
    xpose_in<<<dim3(NF / 2, NN / 32, NR / 32), tblk, 0, stream>>>(X, Xp);
  }
  if (haveW2) {
    sinkhorn_wmma<true, true><<<NR, 256, smem, stream>>>(seed, gum, Xp, ws2);
    xpose_out<<<dim3(NF, NN / 32, NR / 32), tblk, 0, stream>>>(ws2, out);
  } else if (haveXp) {
    sinkhorn_wmma<true, false><<<NR, 256, smem, stream>>>(seed, gum, Xp, out);
  } else {
    sinkhorn_wmma<false, false><<<NR, 256, smem, stream>>>(seed, gum, X, out);
  }
}